// EdgeMLP_81973745812097
// MI455X (gfx1250) — compile-verified
//
#include <hip/hip_runtime.h>
#include <hip/hip_bf16.h>

typedef __attribute__((ext_vector_type(2))) float v2f;
typedef __attribute__((ext_vector_type(8))) float v8f;

#define EDGE_DIM 64
#define NODE_DIM 128
#define LDS_STRIDE 68   // 64 + 4 pad: conflict-free fragment reads

__device__ __forceinline__ float gelu_tanh(float v) {
    // jax.nn.gelu (approximate=True)
    float v3 = v * v * v;
    float t = tanhf(0.7978845608028654f * (v + 0.044715f * v3));
    return 0.5f * v * (1.0f + t);
}

__device__ __forceinline__ void wave_lds_fence() {
    __builtin_amdgcn_wave_barrier();
    asm volatile("s_wait_dscnt 0" ::: "memory");
    __builtin_amdgcn_wave_barrier();
}

// ---------------------------------------------------------------------------
// Kernel 1: node projections  xs = x @ Ws^T + bs ; xt = x @ Wt^T + bt
// One node per 128-thread block; threads 0..63 -> xs, 64..127 -> xt.
// ---------------------------------------------------------------------------
__global__ void node_proj_kernel(const float* __restrict__ x,
                                 const float* __restrict__ Ws,
                                 const float* __restrict__ bs,
                                 const float* __restrict__ Wt,
                                 const float* __restrict__ bt,
                                 float* __restrict__ xs,
                                 float* __restrict__ xt,
                                 int n_nodes) {
    __shared__ float xrow[NODE_DIM];
    int n = blockIdx.x;
    if (n >= n_nodes) return;
    int tid = threadIdx.x;
    xrow[tid] = x[(size_t)n * NODE_DIM + tid];
    __syncthreads();

    const float* W    = (tid < 64) ? Ws : Wt;
    const float* bias = (tid < 64) ? bs : bt;
    float*       o    = (tid < 64) ? xs : xt;
    int j = tid & 63;

    float sum = bias[j];
    const float4* Wr = (const float4*)(W + j * NODE_DIM);
    const float4* xr = (const float4*)xrow;
#pragma unroll
    for (int k = 0; k < NODE_DIM / 4; ++k) {
        float4 w4 = Wr[k];
        float4 x4 = xr[k];
        sum += w4.x * x4.x + w4.y * x4.y + w4.z * x4.z + w4.w * x4.w;
    }
    o[(size_t)n * EDGE_DIM + j] = sum;
}

// ---------------------------------------------------------------------------
// Kernel 2: fused edge MLP using V_WMMA_F32_16X16X4_F32.
// out = gelu(edge_attr@We^T + be + xs[src] + xt[tgt]) @ W1^T + b1
// 8 waves per block; each wave owns one tile of 16 edges.
// ---------------------------------------------------------------------------
__global__ void __launch_bounds__(256)
edge_mlp_kernel(const float* __restrict__ edge_attr,
                const int*   __restrict__ src_idx,   // edge_index[0]
                const int*   __restrict__ tgt_idx,   // edge_index[1]
                const float* __restrict__ xs,
                const float* __restrict__ xt,
                const float* __restrict__ We,
                const float* __restrict__ be,
                const float* __restrict__ W1,
                const float* __restrict__ b1,
                float* __restrict__ out,
                int n_tiles) {
    __shared__ float We_s[EDGE_DIM * LDS_STRIDE];   // B for stage 1 (We^T access)
    __shared__ float W1_s[EDGE_DIM * LDS_STRIDE];   // B for stage 2
    __shared__ float Abuf[8][16 * LDS_STRIDE];      // edge tile / hidden tile
    __shared__ float Gbuf[8][16 * LDS_STRIDE];      // gather+bias tile / output tile

    int tid = threadIdx.x;
    // Stage weights once per block (row-major [64][64] -> stride-68 LDS).
    for (int idx = tid; idx < EDGE_DIM * EDGE_DIM; idx += blockDim.x) {
        int r = idx >> 6, c = idx & 63;
        We_s[r * LDS_STRIDE + c] = We[idx];
        W1_s[r * LDS_STRIDE + c] = W1[idx];
    }
    __syncthreads();

    int w    = __builtin_amdgcn_readfirstlane(tid >> 5);  // wave id (uniform)
    int lane = tid & 31;
    int lr   = lane & 15;       // position within 16-lane half
    int hf   = lane >> 4;       // lane half
    int tile = blockIdx.x * 8 + w;
    if (tile >= n_tiles) return;        // wave-uniform; EXEC stays all-ones

    float* A = Abuf[w];
    float* G = Gbuf[w];
    int e0 = tile * 16;

    // Per-lane constants: be as float2, b1 per N-tile.
    v2f be2 = *(const v2f*)(be + 2 * lane);
    float b1v[4];
#pragma unroll
    for (int t = 0; t < 4; ++t) b1v[t] = b1[16 * t + lr];

    // ---- Stage edge_attr tile and G = be + xs[src] + xt[tgt] into LDS ----
#pragma unroll 4
    for (int m = 0; m < 16; ++m) {
        int e = e0 + m;
        int s = src_idx[e];             // uniform -> scalar loads
        int t = tgt_idx[e];
        v2f ea = *(const v2f*)(edge_attr + (size_t)e * EDGE_DIM + 2 * lane);
        v2f gs = *(const v2f*)(xs + (size_t)s * EDGE_DIM + 2 * lane);
        v2f gt = *(const v2f*)(xt + (size_t)t * EDGE_DIM + 2 * lane);
        *(v2f*)(A + m * LDS_STRIDE + 2 * lane) = ea;
        v2f g; g.x = gs.x + gt.x + be2.x; g.y = gs.y + gt.y + be2.y;
        *(v2f*)(G + m * LDS_STRIDE + 2 * lane) = g;
    }
    wave_lds_fence();

    // ---- Stage 1: acc = G + edge_tile @ We^T  (4 N-tiles of 16x16) ----
    v8f acc0, acc1, acc2, acc3;
#pragma unroll
    for (int r = 0; r < 8; ++r) {
        int m = r + 8 * hf;
        acc0[r] = G[m * LDS_STRIDE +  0 + lr];
        acc1[r] = G[m * LDS_STRIDE + 16 + lr];
        acc2[r] = G[m * LDS_STRIDE + 32 + lr];
        acc3[r] = G[m * LDS_STRIDE + 48 + lr];
    }
#pragma unroll
    for (int kk = 0; kk < 16; ++kk) {
        int k0 = 4 * kk;
        // A fragment: 16x4 fp32 (lane half selects K pair per ISA layout)
        v2f a = *(const v2f*)(A + lr * LDS_STRIDE + k0 + 2 * hf);
        // B fragments: B[k][n] = We[n][k] -> read along We rows
        v2f bB0 = *(const v2f*)(We_s + ( 0 + lr) * LDS_STRIDE + k0 + 2 * hf);
        v2f bB1 = *(const v2f*)(We_s + (16 + lr) * LDS_STRIDE + k0 + 2 * hf);
        v2f bB2 = *(const v2f*)(We_s + (32 + lr) * LDS_STRIDE + k0 + 2 * hf);
        v2f bB3 = *(const v2f*)(We_s + (48 + lr) * LDS_STRIDE + k0 + 2 * hf);
        acc0 = __builtin_amdgcn_wmma_f32_16x16x4_f32(false, a, false, bB0, (short)0, acc0, false, false);
        acc1 = __builtin_amdgcn_wmma_f32_16x16x4_f32(false, a, false, bB1, (short)0, acc1, false, false);
        acc2 = __builtin_amdgcn_wmma_f32_16x16x4_f32(false, a, false, bB2, (short)0, acc2, false, false);
        acc3 = __builtin_amdgcn_wmma_f32_16x16x4_f32(false, a, false, bB3, (short)0, acc3, false, false);
    }

    // ---- GELU, write hidden tile back to LDS (reuse A buffer) ----
#pragma unroll
    for (int r = 0; r < 8; ++r) {
        int m = r + 8 * hf;
        A[m * LDS_STRIDE +  0 + lr] = gelu_tanh(acc0[r]);
        A[m * LDS_STRIDE + 16 + lr] = gelu_tanh(acc1[r]);
        A[m * LDS_STRIDE + 32 + lr] = gelu_tanh(acc2[r]);
        A[m * LDS_STRIDE + 48 + lr] = gelu_tanh(acc3[r]);
    }
    wave_lds_fence();

    // ---- Stage 2: out = b1 + h @ W1^T ----
    v8f o0, o1, o2, o3;
#pragma unroll
    for (int r = 0; r < 8; ++r) { o0[r] = b1v[0]; o1[r] = b1v[1]; o2[r] = b1v[2]; o3[r] = b1v[3]; }
#pragma unroll
    for (int kk = 0; kk < 16; ++kk) {
        int k0 = 4 * kk;
        v2f a = *(const v2f*)(A + lr * LDS_STRIDE + k0 + 2 * hf);
        v2f bB0 = *(const v2f*)(W1_s + ( 0 + lr) * LDS_STRIDE + k0 + 2 * hf);
        v2f bB1 = *(const v2f*)(W1_s + (16 + lr) * LDS_STRIDE + k0 + 2 * hf);
        v2f bB2 = *(const v2f*)(W1_s + (32 + lr) * LDS_STRIDE + k0 + 2 * hf);
        v2f bB3 = *(const v2f*)(W1_s + (48 + lr) * LDS_STRIDE + k0 + 2 * hf);
        o0 = __builtin_amdgcn_wmma_f32_16x16x4_f32(false, a, false, bB0, (short)0, o0, false, false);
        o1 = __builtin_amdgcn_wmma_f32_16x16x4_f32(false, a, false, bB1, (short)0, o1, false, false);
        o2 = __builtin_amdgcn_wmma_f32_16x16x4_f32(false, a, false, bB2, (short)0, o2, false, false);
        o3 = __builtin_amdgcn_wmma_f32_16x16x4_f32(false, a, false, bB3, (short)0, o3, false, false);
    }

    // ---- Write result via LDS for coalesced 8B global stores ----
#pragma unroll
    for (int r = 0; r < 8; ++r) {
        int m = r + 8 * hf;
        G[m * LDS_STRIDE +  0 + lr] = o0[r];
        G[m * LDS_STRIDE + 16 + lr] = o1[r];
        G[m * LDS_STRIDE + 32 + lr] = o2[r];
        G[m * LDS_STRIDE + 48 + lr] = o3[r];
    }
    wave_lds_fence();
#pragma unroll 4
    for (int m = 0; m < 16; ++m) {
        v2f v = *(const v2f*)(G + m * LDS_STRIDE + 2 * lane);
        *(v2f*)(out + (size_t)(e0 + m) * EDGE_DIM + 2 * lane) = v;
    }
}

// ---------------------------------------------------------------------------
// Host entry
// Inputs: 0:x 1:edge_index 2:edge_attr 3:We 4:be 5:Ws 6:bs 7:Wt 8:bt 9:W1 10:b1
// ---------------------------------------------------------------------------
extern "C" void kernel_launch(void* const* d_in, const int* in_sizes, int n_in,
                              void* d_out, int out_size, void* d_ws, size_t ws_size,
                              hipStream_t stream) {
    const float* x         = (const float*)d_in[0];
    const int*   edge_idx  = (const int*)d_in[1];
    const float* edge_attr = (const float*)d_in[2];
    const float* We        = (const float*)d_in[3];
    const float* be        = (const float*)d_in[4];
    const float* Ws        = (const float*)d_in[5];
    const float* bs        = (const float*)d_in[6];
    const float* Wt        = (const float*)d_in[7];
    const float* bt        = (const float*)d_in[8];
    const float* W1        = (const float*)d_in[9];
    const float* b1        = (const float*)d_in[10];
    float*       out       = (float*)d_out;

    int n_nodes = in_sizes[0] / NODE_DIM;
    int n_edges = in_sizes[2] / EDGE_DIM;
    int n_tiles = (n_edges + 15) / 16;   // n_edges = 1M -> 62500 exact

    float* xs = (float*)d_ws;
    float* xt = xs + (size_t)n_nodes * EDGE_DIM;

    node_proj_kernel<<<n_nodes, 128, 0, stream>>>(x, Ws, bs, Wt, bt, xs, xt, n_nodes);

    const int* src = edge_idx;
    const int* tgt = edge_idx + n_edges;
    int blocks = (n_tiles + 7) / 8;
    edge_mlp_kernel<<<blocks, 256, 0, stream>>>(edge_attr, src, tgt, xs, xt,
                                                We, be, W1, b1, out, n_tiles);
}